// GroupedQueryAttention_68693706932753
// MI455X (gfx1250) — compile-verified
//
#include <hip/hip_runtime.h>
#include <math.h>

typedef __attribute__((ext_vector_type(2))) float v2f;
typedef __attribute__((ext_vector_type(8))) float v8f;

#define S_LEN 1024
#define ROWS 64            // row = h*8 + q ; h = row>>3, q = row&7, group g = h>>2
#define CHUNK 128
#define NCHUNK (S_LEN / CHUNK)
#define ATT_STRIDE 132     // CHUNK + 4 pad -> LDS bank stride 4 (conflict-free)
#define SCALE_INV 0.35355339059327373f  // 1/sqrt(HEAD_DIM=8)

// workspace layout (in floats)
#define WS_KSCALE 0
#define WS_VSCALE 16
#define WS_A      32
#define WS_VBASE  96                    // [1024][16]
#define WS_BC     (96 + S_LEN * 16)     // [64 rows][1024 s]

// ---------------------------------------------------------------------------
// P0: kscale/vscale = fe_w @ {k_w,v_w};  A[row] = (Q_slice . kscale)/SCALE
// ---------------------------------------------------------------------------
__global__ void gqa_pre0(const float* __restrict__ queries,
                         const float* __restrict__ fe_w,
                         const float* __restrict__ k_w,
                         const float* __restrict__ v_w,
                         float* __restrict__ ws) {
  __shared__ float ks[16];
  int t = threadIdx.x;  // 64 threads
  if (t < 16) {
    float ak = 0.f, av = 0.f;
#pragma unroll 8
    for (int d = 0; d < 64; ++d) {
      float w = fe_w[d];
      ak = fmaf(w, k_w[d * 16 + t], ak);
      av = fmaf(w, v_w[d * 16 + t], av);
    }
    ks[t] = ak;
    ws[WS_KSCALE + t] = ak;
    ws[WS_VSCALE + t] = av;
  }
  __syncthreads();
  int h = t >> 3, q = t & 7, g = h >> 2;
  float a = 0.f;
#pragma unroll
  for (int d = 0; d < 8; ++d)
    a = fmaf(queries[q * 64 + h * 8 + d], ks[g * 8 + d], a);
  ws[WS_A + t] = a * SCALE_INV;
}

// ---------------------------------------------------------------------------
// P1 (one block per s): k_base/v_base[s] = (fe_b+pos_emb[s]) @ {k_w,v_w} + bias
//                       Bc[row][s] = (Q_slice . k_base[s])/SCALE
// ---------------------------------------------------------------------------
__global__ void gqa_pre1(const float* __restrict__ queries,
                         const float* __restrict__ fe_b,
                         const float* __restrict__ pos_emb,
                         const float* __restrict__ k_w,
                         const float* __restrict__ k_b,
                         const float* __restrict__ v_w,
                         const float* __restrict__ v_b,
                         float* __restrict__ ws) {
  __shared__ float kb[16];
  int s = blockIdx.x;
  int t = threadIdx.x;  // 64 threads
  const float* pe = pos_emb + s * 64;
  if (t < 16) {
    float bk = k_b[t], bv = v_b[t];
#pragma unroll 8
    for (int d = 0; d < 64; ++d) {
      float e = fe_b[d] + pe[d];
      bk = fmaf(e, k_w[d * 16 + t], bk);
      bv = fmaf(e, v_w[d * 16 + t], bv);
    }
    kb[t] = bk;
    ws[WS_VBASE + s * 16 + t] = bv;
  }
  __syncthreads();
  int h = t >> 3, q = t & 7, g = h >> 2;
  float bc = 0.f;
#pragma unroll
  for (int d = 0; d < 8; ++d)
    bc = fmaf(queries[q * 64 + h * 8 + d], kb[g * 8 + d], bc);
  ws[WS_BC + t * S_LEN + s] = bc * SCALE_INV;
}

// ---------------------------------------------------------------------------
// Main: one workgroup (8 waves) per batch b.
//   scores[row][s] = x[b,s]*A[row] + Bc[row][s]
//   softmax over s (online pass), write attn (268 MB -> HBM-bound, b128 stores),
//   ctx = attn @ v_base via V_WMMA_F32_16X16X4_F32 (waves 0-3)
//       + (attn @ x) * vscale  (waves 4-7), then out = ctx @ o_w + o_b.
// ---------------------------------------------------------------------------
__global__ __launch_bounds__(256, 1) void gqa_main(
    const float* __restrict__ x, const float* __restrict__ o_w,
    const float* __restrict__ o_b, const float* __restrict__ ws,
    float* __restrict__ outp, float* __restrict__ attnp) {
  __shared__ float xs[S_LEN];
  __shared__ float mrow[ROWS];
  __shared__ float isum[ROWS];
  __shared__ float attc[ROWS * ATT_STRIDE];
  __shared__ float vch[CHUNK * 16];
  __shared__ float ctxv[ROWS * 16];
  __shared__ float cxr[ROWS];
  __shared__ float ctxm[8 * 64];

  const int b = blockIdx.x;
  const int tid = threadIdx.x;
  const int lane = tid & 31;
  const int wv = tid >> 5;
  const float* Arow = ws + WS_A;
  const float* Bc = ws + WS_BC;
  const float* vbase = ws + WS_VBASE;
  const float* vscale = ws + WS_VSCALE;

  // stage x row (4 KB) into LDS
  ((float4*)xs)[tid] = ((const float4*)(x + (size_t)b * S_LEN))[tid];
  __syncthreads();

  // ---- Pass 1: online softmax stats (each wave owns 8 rows; b128 Bc reads) --
  const float4* xs4 = (const float4*)xs;
#pragma unroll
  for (int rr = 0; rr < 8; ++rr) {
    const int row = wv * 8 + rr;
    const float a = Arow[row];
    const float4* bc4 = (const float4*)(Bc + row * S_LEN);
    float m = -INFINITY, sum = 0.f;
#pragma unroll 2
    for (int i = 0; i < 8; ++i) {
      int si = lane + 32 * i;  // float4 index: covers s = 4*si .. 4*si+3
      float4 bv = bc4[si];
      float4 xv = xs4[si];
      float sc0 = fmaf(xv.x, a, bv.x);
      float sc1 = fmaf(xv.y, a, bv.y);
      float sc2 = fmaf(xv.z, a, bv.z);
      float sc3 = fmaf(xv.w, a, bv.w);
      float cm = fmaxf(fmaxf(sc0, sc1), fmaxf(sc2, sc3));
      float mo = m;
      m = fmaxf(m, cm);
      sum = sum * __expf(mo - m) + __expf(sc0 - m) + __expf(sc1 - m) +
            __expf(sc2 - m) + __expf(sc3 - m);
    }
#pragma unroll
    for (int off = 16; off >= 1; off >>= 1) {
      float m2 = __shfl_xor(m, off);
      float s2 = __shfl_xor(sum, off);
      float mn = fmaxf(m, m2);
      sum = sum * __expf(m - mn) + s2 * __expf(m2 - mn);
      m = mn;
    }
    if (lane == 0) { mrow[row] = m; isum[row] = 1.0f / sum; }
  }
  __syncthreads();

  // ---- Pass 2: chunked attn write + contractions ----
  v8f acc = {0.f, 0.f, 0.f, 0.f, 0.f, 0.f, 0.f, 0.f};  // waves 0-3: WMMA C/D
  float cxp[16];                                        // waves 4-7: sum attn*x
#pragma unroll
  for (int r = 0; r < 16; ++r) cxp[r] = 0.f;

  const size_t abase = (size_t)b * (ROWS * S_LEN);
  const int l16 = lane & 15;
  const int hi = lane >> 4;  // 0: K pair {0,1}/M 0-7 ; 1: K pair {2,3}/M 8-15

  for (int c = 0; c < NCHUNK; ++c) {
    const int s0 = c * CHUNK;
    // stage v_base chunk [128][16] (8 KB)
    ((float4*)vch)[tid] = ((const float4*)(vbase + s0 * 16))[tid];
    ((float4*)vch)[tid + 256] = ((const float4*)(vbase + s0 * 16))[tid + 256];
    // attn for this chunk: each wave covers its 8 rows; lane owns 4 consecutive
    // s values -> one global_store_b128 + one ds_store_b128 per row.
#pragma unroll
    for (int rr = 0; rr < 8; ++rr) {
      const int row = wv * 8 + rr;
      const float a = Arow[row];
      const float m = mrow[row];
      const float is = isum[row];
      const float4 bv = ((const float4*)(Bc + row * S_LEN + s0))[lane];
      const float4 xv = ((const float4*)(xs + s0))[lane];
      float4 p;
      p.x = __expf(fmaf(xv.x, a, bv.x) - m) * is;
      p.y = __expf(fmaf(xv.y, a, bv.y) - m) * is;
      p.z = __expf(fmaf(xv.z, a, bv.z) - m) * is;
      p.w = __expf(fmaf(xv.w, a, bv.w) - m) * is;
      ((float4*)(attnp + abase + (size_t)row * S_LEN + s0))[lane] = p;  // HBM
      ((float4*)(attc + row * ATT_STRIDE))[lane] = p;                   // LDS
    }
    __syncthreads();

    if (wv < 4) {
      // WMMA: M-tile of 16 rows (group-pure), N=16 (g,d) cols, K over chunk
      const int mt = wv;
#pragma unroll
      for (int k0 = 0; k0 < CHUNK; k0 += 4) {
        v2f af, bf;
        const int ka = k0 + 2 * hi;
        af.x = attc[(mt * 16 + l16) * ATT_STRIDE + ka];
        af.y = attc[(mt * 16 + l16) * ATT_STRIDE + ka + 1];
        bf.x = vch[ka * 16 + l16];
        bf.y = vch[(ka + 1) * 16 + l16];
        acc = __builtin_amdgcn_wmma_f32_16x16x4_f32(false, af, false, bf,
                                                    (short)0, acc, false, false);
      }
    } else {
      const int rbase = (wv - 4) * 16;
#pragma unroll
      for (int r = 0; r < 16; ++r) {
        const int row = rbase + r;
        float t = 0.f;
#pragma unroll
        for (int i = 0; i < 4; ++i) {
          int sl = lane + 32 * i;
          t = fmaf(attc[row * ATT_STRIDE + sl], xs[s0 + sl], t);
        }
        cxp[r] += t;
      }
    }
    __syncthreads();
  }

  // ---- unload accumulators ----
  if (wv < 4) {
    // D layout: VGPR r -> M = r (lanes 0-15) / r+8 (lanes 16-31), N = lane&15
#pragma unroll
    for (int r = 0; r < 8; ++r)
      ctxv[(wv * 16 + hi * 8 + r) * 16 + l16] = acc[r];
  } else {
    const int rbase = (wv - 4) * 16;
#pragma unroll
    for (int r = 0; r < 16; ++r) {
      float t = cxp[r];
#pragma unroll
      for (int off = 16; off >= 1; off >>= 1) t += __shfl_xor(t, off);
      if (lane == 0) cxr[rbase + r] = t;
    }
  }
  __syncthreads();

  // ---- ctx[q][i]  (i = h*8+d; row = h*8+q; keep group-g columns) ----
  for (int e = tid; e < 512; e += 256) {
    int q = e >> 6, i = e & 63;
    int h = i >> 3, d = i & 7, g = h >> 2;
    int row = h * 8 + q;
    ctxm[q * 64 + i] = fmaf(cxr[row], vscale[g * 8 + d], ctxv[row * 16 + g * 8 + d]);
  }
  __syncthreads();

  // ---- out = ctx @ o_w + o_b   ([8,64]@[64,64], tiny) ----
  for (int e = tid; e < 512; e += 256) {
    int q = e >> 6, j = e & 63;
    float t = o_b[j];
#pragma unroll 8
    for (int i = 0; i < 64; ++i)
      t = fmaf(ctxm[q * 64 + i], o_w[i * 64 + j], t);
    outp[(size_t)b * 512 + q * 64 + j] = t;
  }
}

// ---------------------------------------------------------------------------
extern "C" void kernel_launch(void* const* d_in, const int* in_sizes, int n_in,
                              void* d_out, int out_size, void* d_ws,
                              size_t ws_size, hipStream_t stream) {
  const float* x       = (const float*)d_in[0];
  const float* queries = (const float*)d_in[1];
  const float* fe_w    = (const float*)d_in[2];
  const float* fe_b    = (const float*)d_in[3];
  const float* pos_emb = (const float*)d_in[4];
  const float* k_w     = (const float*)d_in[5];
  const float* k_b     = (const float*)d_in[6];
  const float* v_w     = (const float*)d_in[7];
  const float* v_b     = (const float*)d_in[8];
  const float* o_w     = (const float*)d_in[9];
  const float* o_b     = (const float*)d_in[10];

  float* outp  = (float*)d_out;              // [1024, 512]
  float* attnp = outp + 1024 * 512;          // [1024, 8, 8, 1024]
  float* ws    = (float*)d_ws;               // ~328 KB used

  gqa_pre0<<<1, 64, 0, stream>>>(queries, fe_w, k_w, v_w, ws);
  gqa_pre1<<<1024, 64, 0, stream>>>(queries, fe_b, pos_emb, k_w, k_b, v_w, v_b, ws);
  gqa_main<<<1024, 256, 0, stream>>>(x, o_w, o_b, ws, outp, attnp);
}